// GraphConvolutionalNetwork_89635967467589
// MI455X (gfx1250) — compile-verified
//
#include <hip/hip_runtime.h>
#include <hip/hip_bf16.h>
#include <math.h>

typedef float v2f __attribute__((ext_vector_type(2)));
typedef float v8f __attribute__((ext_vector_type(8)));

#define HDIM 128
#define LDSA_STRIDE 132   // 128 + 4 pad -> conflict-free column reads, keeps 16B align

// ---------------------------------------------------------------------------
// Dense GEMM  XW = Hin @ W   (N x 128) @ (128 x 128), fp32 WMMA 16x16x4.
// Block = 256 threads = 8 waves; wave w owns output columns [16w, 16w+16).
// B slice held in 64 VGPRs per wave; A tiles double-buffered in LDS and
// prefetched with GLOBAL_LOAD_ASYNC_TO_LDS_B128 (ASYNCcnt) to overlap the
// next tile's fetch with the current tile's WMMAs.
// ---------------------------------------------------------------------------
__device__ __forceinline__ void prefetch_tile_async(
    const float* __restrict__ Hin, int rowBase, float* lbuf, int tid)
{
#pragma unroll
    for (int i = 0; i < 2; ++i) {
        const int q   = tid + i * 256;      // float4 index within 16x128 tile
        const int row = q >> 5;             // (q*4)/128
        const int col = (q & 31) * 4;       // (q*4)%128
        const unsigned int  loff  = (unsigned int)(size_t)(lbuf + row * LDSA_STRIDE + col);
        const unsigned long gaddr = (unsigned long)(size_t)(Hin + (size_t)rowBase * HDIM + q * 4);
        asm volatile("global_load_async_to_lds_b128 %0, %1, off"
                     :: "v"(loff), "v"(gaddr) : "memory");
    }
}

__global__ __launch_bounds__(256) void gcn_gemm_wmma(
    const float* __restrict__ Hin, const float* __restrict__ W,
    float* __restrict__ XW, int nrows)
{
    __shared__ float ldsA[2][16 * LDSA_STRIDE];

    const int tid     = threadIdx.x;
    const int lane    = tid & 31;
    const int wave    = tid >> 5;              // 0..7 column tile
    const int r16     = lane & 15;
    const int koff    = (lane >> 4) * 2;       // 0 (lanes 0-15) or 2 (lanes 16-31)
    const int col     = wave * 16 + r16;       // output column this lane covers
    const int rowAdd  = (lane >> 4) * 8;       // C layout: high lanes hold M+8

    // Preload this wave's B slice (128 x 16) into registers: 32 k-steps of v2f.
    v2f breg[32];
#pragma unroll
    for (int kk = 0; kk < 32; ++kk) {
        const int k = kk * 4 + koff;
        breg[kk].x = W[k * HDIM + col];
        breg[kk].y = W[(k + 1) * HDIM + col];
    }

    const int numTiles = nrows >> 4;           // N = 50000 -> 3125 tiles
    int buf = 0;

    if ((int)blockIdx.x < numTiles)
        prefetch_tile_async(Hin, (int)blockIdx.x << 4, &ldsA[0][0], tid);

    for (int t = blockIdx.x; t < numTiles; t += gridDim.x) {
        const int tn = t + gridDim.x;
        if (tn < numTiles) {
            // start fetching the next tile into the other buffer, then wait
            // only for the *current* tile's 2 async ops (in-order completion)
            prefetch_tile_async(Hin, tn << 4, &ldsA[buf ^ 1][0], tid);
            asm volatile("s_wait_asynccnt 0x2" ::: "memory");
        } else {
            asm volatile("s_wait_asynccnt 0x0" ::: "memory");
        }
        __syncthreads();                        // all waves' tile data visible

        const float* __restrict__ A = &ldsA[buf][0];
        v8f acc = {0.f, 0.f, 0.f, 0.f, 0.f, 0.f, 0.f, 0.f};
#pragma unroll
        for (int kk = 0; kk < 32; ++kk) {
            const int k = kk * 4 + koff;
            v2f a;
            a.x = A[r16 * LDSA_STRIDE + k];
            a.y = A[r16 * LDSA_STRIDE + k + 1];
            acc = __builtin_amdgcn_wmma_f32_16x16x4_f32(
                false, a, false, breg[kk], (short)0, acc, false, false);
        }

        const int rowBase = t << 4;
#pragma unroll
        for (int j = 0; j < 8; ++j) {
            XW[(size_t)(rowBase + j + rowAdd) * HDIM + col] = acc[j];
        }
        __syncthreads();                        // reads done; buffer may be reused
        buf ^= 1;
    }
}

// ---------------------------------------------------------------------------
// Small utility / graph kernels
// ---------------------------------------------------------------------------
__global__ void fill_f32(float* __restrict__ p, float v, int n) {
    int i = blockIdx.x * blockDim.x + threadIdx.x;
    if (i < n) p[i] = v;
}

__global__ void copy_f32(float* __restrict__ dst, const float* __restrict__ src, int n) {
    int i = blockIdx.x * blockDim.x + threadIdx.x;
    if (i < n) dst[i] = src[i];
}

__global__ void edge_deg_kernel(const int* __restrict__ dst, float* __restrict__ deg, int ne) {
    int e = blockIdx.x * blockDim.x + threadIdx.x;
    if (e < ne) atomicAdd(&deg[dst[e]], 1.0f);
}

__global__ void rsqrt_kernel(const float* __restrict__ deg, float* __restrict__ dis, int n) {
    int i = blockIdx.x * blockDim.x + threadIdx.x;
    if (i < n) dis[i] = rsqrtf(deg[i]);
}

__global__ void edge_norm_kernel(const int* __restrict__ src, const int* __restrict__ dst,
                                 const float* __restrict__ dis, float* __restrict__ nrm, int ne) {
    int e = blockIdx.x * blockDim.x + threadIdx.x;
    if (e < ne) nrm[e] = dis[src[e]] * dis[dst[e]];
}

// One wave (32 lanes) per edge; each lane moves 4 consecutive floats (float4).
__global__ __launch_bounds__(256) void edge_scatter_kernel(
    const int* __restrict__ src, const int* __restrict__ dst,
    const float* __restrict__ nrm, const float* __restrict__ XW,
    float* __restrict__ AGG, int ne)
{
    int gid = blockIdx.x * blockDim.x + threadIdx.x;
    int e = gid >> 5;
    if (e >= ne) return;
    int lane = gid & 31;
    int s = src[e], d = dst[e];
    float w = nrm[e];
    const float4 v = *(const float4*)(XW + (size_t)s * HDIM + lane * 4);
    float* out = AGG + (size_t)d * HDIM + lane * 4;
    atomicAdd(out + 0, v.x * w);
    atomicAdd(out + 1, v.y * w);
    atomicAdd(out + 2, v.z * w);
    atomicAdd(out + 3, v.w * w);
}

// h = relu(agg + xw*dis^2 + bias) + h   (self-loop term folded in here)
__global__ void layer_fuse_kernel(const float* __restrict__ AGG,
                                  const float* __restrict__ XW,
                                  const float* __restrict__ dis,
                                  const float* __restrict__ bias,
                                  float* __restrict__ Hbuf, int n)
{
    int i = blockIdx.x * blockDim.x + threadIdx.x;
    if (i >= n) return;
    int node = i >> 7;
    int f = i & 127;
    float d = dis[node];
    float v = AGG[i] + XW[i] * d * d + bias[f];
    Hbuf[i] = fmaxf(v, 0.0f) + Hbuf[i];
}

// Segment max/sum pooling via atomics (max uses the signed/unsigned bit trick).
__global__ void pool_kernel(const float* __restrict__ Hbuf, const int* __restrict__ batch,
                            float* __restrict__ maxp, float* __restrict__ addp, int n)
{
    int i = blockIdx.x * blockDim.x + threadIdx.x;
    if (i >= n) return;
    int node = i >> 7;
    int f = i & 127;
    int g = batch[node];
    float v = Hbuf[i];
    atomicAdd(&addp[g * HDIM + f], v);
    float* a = &maxp[g * HDIM + f];
    if (v >= 0.0f) atomicMax((int*)a, __float_as_int(v));
    else           atomicMin((unsigned int*)a, __float_as_uint(v));
}

// mol = relu((mf @ Wm + bm) * gamma + beta)   [256 x 32], K = 200
__global__ void mol_kernel(const float* __restrict__ mf, const float* __restrict__ Wm,
                           const float* __restrict__ bm, const float* __restrict__ gamma,
                           const float* __restrict__ beta, float* __restrict__ mol, int n)
{
    int i = blockIdx.x * blockDim.x + threadIdx.x;
    if (i >= n) return;
    int b = i >> 5;
    int j = i & 31;
    float acc = bm[j];
    for (int k = 0; k < 200; ++k) acc += mf[b * 200 + k] * Wm[k * 32 + j];
    acc = acc * gamma[j] + beta[j];
    mol[i] = fmaxf(acc, 0.0f);
}

// z1 = relu(concat(maxp, addp, mol) @ W1 + b1)   [256 x 128], K = 288
__global__ void mlp1_kernel(const float* __restrict__ maxp, const float* __restrict__ addp,
                            const float* __restrict__ mol, const float* __restrict__ W1,
                            const float* __restrict__ b1, float* __restrict__ z1, int n)
{
    int i = blockIdx.x * blockDim.x + threadIdx.x;
    if (i >= n) return;
    int b = i >> 7;
    int h = i & 127;
    float acc = b1[h];
    for (int k = 0; k < 128; ++k) acc += maxp[b * HDIM + k] * W1[k * HDIM + h];
    for (int k = 0; k < 128; ++k) acc += addp[b * HDIM + k] * W1[(128 + k) * HDIM + h];
    for (int k = 0; k < 32;  ++k) acc += mol[b * 32 + k]    * W1[(256 + k) * HDIM + h];
    z1[i] = fmaxf(acc, 0.0f);
}

// out = z1 @ W2 + b2   [256 x 12], K = 128
__global__ void mlp2_kernel(const float* __restrict__ z1, const float* __restrict__ W2,
                            const float* __restrict__ b2, float* __restrict__ out, int n)
{
    int i = blockIdx.x * blockDim.x + threadIdx.x;
    if (i >= n) return;
    int b = i / 12;
    int o = i % 12;
    float acc = b2[o];
    for (int k = 0; k < 128; ++k) acc += z1[b * HDIM + k] * W2[k * 12 + o];
    out[i] = acc;
}

// ---------------------------------------------------------------------------
extern "C" void kernel_launch(void* const* d_in, const int* in_sizes, int n_in,
                              void* d_out, int out_size, void* d_ws, size_t ws_size,
                              hipStream_t stream)
{
    (void)in_sizes; (void)n_in; (void)out_size; (void)ws_size;

    const float* x     = (const float*)d_in[0];
    const int*   eidx  = (const int*)d_in[1];
    const int*   batch = (const int*)d_in[2];
    const float* mf    = (const float*)d_in[3];
    const float* Ws    = (const float*)d_in[4];
    const float* bs    = (const float*)d_in[5];
    const float* Wm    = (const float*)d_in[6];
    const float* bm    = (const float*)d_in[7];
    const float* gamma = (const float*)d_in[8];
    const float* beta  = (const float*)d_in[9];
    const float* W1    = (const float*)d_in[10];
    const float* b1    = (const float*)d_in[11];
    const float* W2    = (const float*)d_in[12];
    const float* b2    = (const float*)d_in[13];

    constexpr int N = 50000, E = 600000, H = 128, B = 256, MH = 32, OUT = 12, L = 4;
    const int* srcIdx = eidx;
    const int* dstIdx = eidx + E;

    float* ws   = (float*)d_ws;
    float* hbuf = ws; ws += (size_t)N * H;
    float* xw   = ws; ws += (size_t)N * H;
    float* agg  = ws; ws += (size_t)N * H;
    float* deg  = ws; ws += N;
    float* dis  = ws; ws += N;
    float* nrm  = ws; ws += E;
    float* maxp = ws; ws += B * H;
    float* addp = ws; ws += B * H;
    float* mol  = ws; ws += B * MH;
    float* z1   = ws; ws += B * H;

    const int NH = N * H;
    auto g = [](int n) { return (n + 255) / 256; };

    // Degrees (self loop = 1.0 init), dis = rsqrt(deg), per-edge norm (layer-invariant).
    fill_f32<<<g(N), 256, 0, stream>>>(deg, 1.0f, N);
    edge_deg_kernel<<<g(E), 256, 0, stream>>>(dstIdx, deg, E);
    rsqrt_kernel<<<g(N), 256, 0, stream>>>(deg, dis, N);
    edge_norm_kernel<<<g(E), 256, 0, stream>>>(srcIdx, dstIdx, dis, nrm, E);

    // h = x (inputs must not be mutated)
    copy_f32<<<g(NH), 256, 0, stream>>>(hbuf, x, NH);

    for (int l = 0; l < L; ++l) {
        gcn_gemm_wmma<<<768, 256, 0, stream>>>(hbuf, Ws + (size_t)l * H * H, xw, N);
        fill_f32<<<g(NH), 256, 0, stream>>>(agg, 0.0f, NH);
        edge_scatter_kernel<<<g(E * 32), 256, 0, stream>>>(srcIdx, dstIdx, nrm, xw, agg, E);
        layer_fuse_kernel<<<g(NH), 256, 0, stream>>>(agg, xw, dis, bs + l * H, hbuf, NH);
    }

    // Pooling
    fill_f32<<<g(B * H), 256, 0, stream>>>(maxp, -INFINITY, B * H);
    fill_f32<<<g(B * H), 256, 0, stream>>>(addp, 0.0f, B * H);
    pool_kernel<<<g(NH), 256, 0, stream>>>(hbuf, batch, maxp, addp, NH);

    // Heads
    mol_kernel<<<g(B * MH), 256, 0, stream>>>(mf, Wm, bm, gamma, beta, mol, B * MH);
    mlp1_kernel<<<g(B * H), 256, 0, stream>>>(maxp, addp, mol, W1, b1, z1, B * H);
    mlp2_kernel<<<g(B * OUT), 256, 0, stream>>>(z1, W2, b2, (float*)d_out, B * OUT);
}